// MSA1_7421703487730
// MI455X (gfx1250) — compile-verified
//
#include <hip/hip_runtime.h>
#include <hip/hip_bf16.h>

#define BATCH 4
#define NHID  128
#define L     1024
#define NH    10
#define HD    10
#define DQ    100           // NH*HD
#define RET_SIZE (BATCH*NHID*L)   // 524288

typedef float v2f __attribute__((ext_vector_type(2)));
typedef float v8f __attribute__((ext_vector_type(8)));

// ---------------- Phase 1: q = Wq @ x + bq  (B,100,1024) ----------------
__global__ __launch_bounds__(256) void msa_q_kernel(
    const float* __restrict__ x, const float* __restrict__ wq_w,
    const float* __restrict__ wq_b, float* __restrict__ q)
{
    int idx = blockIdx.x * 256 + threadIdx.x;          // B*DQ*L = 409600
    if (idx >= BATCH * DQ * L) return;
    int l = idx % L;
    int c = (idx / L) % DQ;
    int b = idx / (L * DQ);
    const float* xb = x + (size_t)b * NHID * L + l;    // stride L over h
    const float* w  = wq_w + c * NHID;
    float s = wq_b[c];
#pragma unroll 4
    for (int h = 0; h < NHID; ++h) s = fmaf(w[h], xb[(size_t)h * L], s);
    q[idx] = s;
}

// ---------------- Phase 1b: s_i / s_j projections ----------------
__global__ __launch_bounds__(256) void msa_s_kernel(
    const float* __restrict__ q, const float* __restrict__ wedge_w,
    float* __restrict__ s_i, float* __restrict__ s_j)
{
    int idx = blockIdx.x * 256 + threadIdx.x;          // B*NH*L = 40960
    if (idx >= BATCH * NH * L) return;
    int l  = idx % L;
    int bn = idx / L;
    int b  = bn / NH, n = bn % NH;
    const float* qp = q + ((size_t)b * DQ + n * HD) * L + l;
    float a = 0.f, c2 = 0.f;
#pragma unroll
    for (int d = 0; d < HD; ++d) {
        float kv = qp[(size_t)d * L];
        a  = fmaf(kv, wedge_w[d], a);
        c2 = fmaf(kv, wedge_w[HD + d], c2);
    }
    s_i[idx] = a;
    s_j[idx] = c2;
}

// ---------------- Phase 2: fused mask + online-softmax + alpha@k (WMMA f32) ----------------
// grid = B*NH*(L/16) = 2560 workgroups, 256 threads (8 waves).
__global__ __launch_bounds__(256) void msa_attn_kernel(
    const float* __restrict__ q, const float* __restrict__ s_i,
    const float* __restrict__ s_j, const int* __restrict__ edge,
    const float* __restrict__ wedge_b,
    float* __restrict__ alpha_out, float* __restrict__ att_out)
{
    const int tile = blockIdx.x;
    const int bn   = tile / (L / 16);                  // b*NH + n
    const int i0   = (tile % (L / 16)) * 16;
    const int b    = bn / NH;
    const int n    = bn % NH;

    const int t    = threadIdx.x;
    const int wave = t >> 5;                           // 0..7
    const int lane = t & 31;

    __shared__ __align__(16) float sj_lds[L];          // staged s_j row (4 KB)
    __shared__ float si_lds[16];
    __shared__ float p_lds[8][16 * 68];                // per-wave 16x64 p tiles (stride 68)
    __shared__ float redm[16][16];
    __shared__ float reds[16][16];
    __shared__ float rowmax[16];
    __shared__ float rowsum[16];
    __shared__ float att_lds[8][16][16];

    const float bias = wedge_b[0];
    const float inv  = 0.31622776601683794f;           // 1/sqrt(HD)

    // ---- async-stage s_j row into LDS (16 B per lane, 256 lanes = 4 KB) ----
    {
        unsigned      ldsoff = (unsigned)(uintptr_t)(&sj_lds[t * 4]);
        const float*  gsrc   = s_j + (size_t)bn * L + t * 4;
        asm volatile("global_load_async_to_lds_b128 %0, %1, off"
                     :: "v"(ldsoff), "v"(gsrc) : "memory");
    }
    if (t < 16) si_lds[t] = s_i[bn * L + i0 + t];
    __builtin_prefetch(edge + ((size_t)b * L + i0) * L, 0, 1);
    asm volatile("s_wait_asynccnt 0x0" ::: "memory");
    __syncthreads();

    // ---- fused pass: online row max + sum over masked scores ----
    {
        int row = t >> 4, sub = t & 15;
        float si = si_lds[row];
        const int* erow = edge + ((size_t)b * L + (i0 + row)) * L;
        float m = -3.0e38f, sum = 0.f;
        for (int j = sub; j < L; j += 16) {
            float v = (si + sj_lds[j] + bias) * inv;
            v = (erow[j] == 0) ? -1.0e15f : v;
            if (v > m) {
                sum = sum * __expf(m - v) + 1.0f;
                m = v;
            } else {
                sum += __expf(v - m);
            }
        }
        redm[row][sub] = m;
        reds[row][sub] = sum;
    }
    __syncthreads();
    if (t < 16) {
        float M = redm[t][0];
#pragma unroll
        for (int s = 1; s < 16; ++s) M = fmaxf(M, redm[t][s]);
        float S = 0.f;
#pragma unroll
        for (int s = 0; s < 16; ++s) S += reds[t][s] * __expf(redm[t][s] - M);
        rowmax[t] = M;
        rowsum[t] = S;
    }
    __syncthreads();

    // ---- pass C: per-wave 128-wide j-chunk: p -> LDS + alpha(NT), then WMMA att += p@k ----
    const int jc = wave * 128;
    float* pw = &p_lds[wave][0];
    v8f acc = {};

    for (int half = 0; half < 2; ++half) {
        const int jh = jc + half * 64;
        // fill 16x64 p tile (each lane: 16 rows x 2 cols)
        for (int row = 0; row < 16; ++row) {
            float si = si_lds[row];
            float m  = rowmax[row];
            float rs = rowsum[row];
            const int* erow = edge + ((size_t)b * L + (i0 + row)) * L + jh;
            float*    arow  = alpha_out + ((size_t)bn * L + (i0 + row)) * L + jh;
#pragma unroll
            for (int c = lane; c < 64; c += 32) {
                float v = (si + sj_lds[jh + c] + bias) * inv;
                v = (erow[c] == 0) ? -1.0e15f : v;
                float p = __expf(v - m) / rs;
                pw[row * 68 + c] = p;
                __builtin_nontemporal_store(p, arow + c);   // alpha never re-read: keep L2 for edge
            }
        }
        // WMMA sweep over this 64-wide chunk, K steps of 4
        const int M    = lane & 15;
        const int koff = (lane >> 4) * 2;              // 0 or 2
        const int dN   = lane & 15;                    // N index (head dim, padded)
        for (int j0 = 0; j0 < 64; j0 += 4) {
            v2f a, bf;
            a.x = pw[M * 68 + j0 + koff];
            a.y = pw[M * 68 + j0 + koff + 1];
            bf.x = 0.f; bf.y = 0.f;
            if (dN < HD) {
                const float* qcol = q + ((size_t)b * DQ + n * HD + dN) * L + jh + j0 + koff;
                bf.x = qcol[0];
                bf.y = qcol[1];
            }
            acc = __builtin_amdgcn_wmma_f32_16x16x4_f32(
                false, a, false, bf, (short)0, acc, false, false);
        }
    }

    // ---- spill per-wave accumulators and reduce ----
    {
        const int Nn    = lane & 15;
        const int mbase = (lane >> 4) * 8;             // C/D layout: vgpr r -> M = r + 0/8
#pragma unroll
        for (int r = 0; r < 8; ++r)
            att_lds[wave][mbase + r][Nn] = acc[r];
    }
    __syncthreads();
    if (t < 16 * HD) {
        int row = t / HD;
        int d   = t % HD;
        float s = 0.f;
#pragma unroll
        for (int w = 0; w < 8; ++w) s += att_lds[w][row][d];
        att_out[(size_t)bn * L * HD + (size_t)(i0 + row) * HD + d] = s;
    }
}

// ---------------- Phase 3: ret = Wo @ att_view + bo ----------------
__global__ __launch_bounds__(256) void msa_out_kernel(
    const float* __restrict__ att, const float* __restrict__ wo_w,
    const float* __restrict__ wo_b, float* __restrict__ ret)
{
    int idx = blockIdx.x * 256 + threadIdx.x;          // B*NHID*L = 524288
    if (idx >= BATCH * NHID * L) return;
    int l = idx % L;
    int o = (idx / L) % NHID;
    int b = idx / (L * NHID);
    const float* a = att + (size_t)b * DQ * L + l;     // scrambled (B,100,1024) view
    const float* w = wo_w + o * DQ;
    float s = wo_b[o];
#pragma unroll 4
    for (int c = 0; c < DQ; ++c) s = fmaf(w[c], a[(size_t)c * L], s);
    ret[idx] = s;
}

extern "C" void kernel_launch(void* const* d_in, const int* in_sizes, int n_in,
                              void* d_out, int out_size, void* d_ws, size_t ws_size,
                              hipStream_t stream) {
    const float* x       = (const float*)d_in[0];
    const int*   edge    = (const int*)  d_in[1];
    const float* wq_w    = (const float*)d_in[2];
    const float* wq_b    = (const float*)d_in[3];
    const float* wedge_w = (const float*)d_in[4];
    const float* wedge_b = (const float*)d_in[5];
    const float* wo_w    = (const float*)d_in[6];
    const float* wo_b    = (const float*)d_in[7];

    float* out       = (float*)d_out;
    float* ret       = out;                    // (B,128,1024,1)
    float* alpha_out = out + RET_SIZE;         // (B*NH, L, L)

    float* ws  = (float*)d_ws;
    float* q   = ws;                           // 409600
    float* s_i = q   + (size_t)BATCH * DQ * L; // 40960
    float* s_j = s_i + (size_t)BATCH * NH * L; // 40960
    float* att = s_j + (size_t)BATCH * NH * L; // 409600

    msa_q_kernel   <<<(BATCH * DQ * L) / 256, 256, 0, stream>>>(x, wq_w, wq_b, q);
    msa_s_kernel   <<<(BATCH * NH * L) / 256, 256, 0, stream>>>(q, wedge_w, s_i, s_j);
    msa_attn_kernel<<<BATCH * NH * (L / 16), 256, 0, stream>>>(q, s_i, s_j, edge,
                                                               wedge_b, alpha_out, att);
    msa_out_kernel <<<(BATCH * NHID * L) / 256, 256, 0, stream>>>(att, wo_w, wo_b, ret);
}